// L2BTAnomalyMapGenerator_7258494730328
// MI455X (gfx1250) — compile-verified
//
#include <hip/hip_runtime.h>

typedef float v2f __attribute__((ext_vector_type(2)));
typedef float v8f __attribute__((ext_vector_type(8)));

#define B_    16
#define N_    1024
#define C_    768
#define HP_   32
#define H_    448
#define NPIX  (H_ * H_)
#define TOPK  200
#define EPSN  1e-12f

// ---------------------------------------------------------------------------
// Kernel 1: per-patch anomaly distances, combined = dist_mid * dist_last
// One wave (32 lanes) per patch; 768 floats/tensor -> 6 x float4 per lane.
// ---------------------------------------------------------------------------
__global__ void anomaly_combined_kernel(const float* __restrict__ mid,
                                        const float* __restrict__ lst,
                                        const float* __restrict__ pmid,
                                        const float* __restrict__ plst,
                                        float* __restrict__ comb) {
    int wave = (blockIdx.x * blockDim.x + threadIdx.x) >> 5;   // 0..16383
    int lane = threadIdx.x & 31;
    size_t base = (size_t)wave * C_;
    const float4* m4 = (const float4*)(mid  + base);
    const float4* l4 = (const float4*)(lst  + base);
    const float4* p4 = (const float4*)(pmid + base);
    const float4* q4 = (const float4*)(plst + base);

    float smm = 0.f, spm = 0.f, dm = 0.f;
    float sll = 0.f, spl = 0.f, dl = 0.f;
#pragma unroll
    for (int i = 0; i < 6; ++i) {
        float4 a = m4[i * 32 + lane];
        float4 p = p4[i * 32 + lane];
        smm += a.x * a.x + a.y * a.y + a.z * a.z + a.w * a.w;
        spm += p.x * p.x + p.y * p.y + p.z * p.z + p.w * p.w;
        dm  += a.x * p.x + a.y * p.y + a.z * p.z + a.w * p.w;
        float4 b = l4[i * 32 + lane];
        float4 q = q4[i * 32 + lane];
        sll += b.x * b.x + b.y * b.y + b.z * b.z + b.w * b.w;
        spl += q.x * q.x + q.y * q.y + q.z * q.z + q.w * q.w;
        dl  += b.x * q.x + b.y * q.y + b.z * q.z + b.w * q.w;
    }
#pragma unroll
    for (int off = 16; off > 0; off >>= 1) {
        smm += __shfl_xor(smm, off, 32);
        spm += __shfl_xor(spm, off, 32);
        dm  += __shfl_xor(dm,  off, 32);
        sll += __shfl_xor(sll, off, 32);
        spl += __shfl_xor(spl, off, 32);
        dl  += __shfl_xor(dl,  off, 32);
    }
    if (lane == 0) {
        float nm  = fmaxf(sqrtf(smm), EPSN);
        float npm = fmaxf(sqrtf(spm), EPSN);
        float d2m = smm / (nm * nm) + spm / (npm * npm) - 2.f * dm / (nm * npm);
        float distm = sqrtf(fmaxf(d2m, 0.f));
        float nl  = fmaxf(sqrtf(sll), EPSN);
        float npl = fmaxf(sqrtf(spl), EPSN);
        float d2l = sll / (nl * nl) + spl / (npl * npl) - 2.f * dl / (nl * npl);
        float distl = sqrtf(fmaxf(d2l, 0.f));
        comb[wave] = distm * distl;
    }
}

// ---------------------------------------------------------------------------
// Kernel 2: bilinear-upsample matrix U (448x32), half-pixel centers, edge clamp
// ---------------------------------------------------------------------------
__global__ void compute_u_kernel(float* __restrict__ U) {
    int r = blockIdx.x;      // 0..447
    int j = threadIdx.x;     // 0..31
    float src  = (r + 0.5f) / 14.0f - 0.5f;
    float f    = floorf(src);
    int   j0   = (int)f;
    float frac = src - f;
    int c0 = min(max(j0, 0), HP_ - 1);
    int c1 = min(max(j0 + 1, 0), HP_ - 1);
    float w = 0.f;
    if (c0 == j) w += 1.f - frac;
    if (c1 == j) w += frac;
    U[r * HP_ + j] = w;
}

// ---------------------------------------------------------------------------
// Kernel 3: 1-D box blur (zero padded) applied down the 448 dim of a 448x32 mat
// ---------------------------------------------------------------------------
__global__ void blur1d_kernel(const float* __restrict__ in,
                              float* __restrict__ out, int W) {
    int r = blockIdx.x;     // 0..447
    int j = threadIdx.x;    // 0..31
    int pad = W >> 1;
    float s = 0.f;
    for (int t = -pad; t <= pad; ++t) {
        int rr = r + t;
        if (rr >= 0 && rr < H_) s += in[rr * HP_ + j];
    }
    out[r * HP_ + j] = s / (float)W;
}

// ---------------------------------------------------------------------------
// WMMA GEMM 1: T[b] (448x32) = M (448x32) x A[b] (32x32), fp32 16x16x4 WMMA.
// One wave per 16x16 output tile. 16 batches * 28 mtiles * 2 ntiles = 896 waves
// ---------------------------------------------------------------------------
__global__ void gemm1_kernel(const float* __restrict__ M,
                             const float* __restrict__ comb,
                             float* __restrict__ T) {
    int wid  = (blockIdx.x * blockDim.x + threadIdx.x) >> 5;   // 0..895
    int lane = threadIdx.x & 31;
    int b  = wid / 56;
    int t  = wid % 56;
    int mt = t >> 1;         // 0..27
    int nt = t & 1;          // 0..1
    const float* A = comb + b * N_;   // 32x32 row-major (rows = y)

    int m  = lane & 15;
    int kb = (lane >> 4) << 1;

    v8f acc = {0.f, 0.f, 0.f, 0.f, 0.f, 0.f, 0.f, 0.f};
#pragma unroll
    for (int k = 0; k < 32; k += 4) {
        v2f a = *(const v2f*)(M + (mt * 16 + m) * HP_ + k + kb);
        v2f bf;
        bf.x = A[(k + kb)     * HP_ + nt * 16 + m];
        bf.y = A[(k + kb + 1) * HP_ + nt * 16 + m];
        acc = __builtin_amdgcn_wmma_f32_16x16x4_f32(false, a, false, bf,
                                                    (short)0, acc, false, false);
    }
    float* Tb = T + (size_t)b * (H_ * HP_);
    int rowoff = (lane >> 4) << 3;
#pragma unroll
    for (int i = 0; i < 8; ++i)
        Tb[(mt * 16 + i + rowoff) * HP_ + nt * 16 + m] = acc[i];
}

// ---------------------------------------------------------------------------
// WMMA GEMM 2: Out[b] (448x448) = T[b] (448x32) x M^T (32x448).
// B fragment rows (k,k+1) at fixed col n are contiguous in M -> float2 load.
// 16 batches * 28 * 28 tiles = 12544 waves.
// ---------------------------------------------------------------------------
__global__ void gemm2_kernel(const float* __restrict__ T,
                             const float* __restrict__ M,
                             float* __restrict__ out) {
    int wid  = (blockIdx.x * blockDim.x + threadIdx.x) >> 5;   // 0..12543
    int lane = threadIdx.x & 31;
    int b  = wid / 784;
    int t  = wid % 784;
    int mt = t / 28;
    int nt = t % 28;
    const float* Tb = T + (size_t)b * (H_ * HP_);

    int m  = lane & 15;
    int kb = (lane >> 4) << 1;

    v8f acc = {0.f, 0.f, 0.f, 0.f, 0.f, 0.f, 0.f, 0.f};
#pragma unroll
    for (int k = 0; k < 32; k += 4) {
        v2f a  = *(const v2f*)(Tb + (mt * 16 + m) * HP_ + k + kb);
        v2f bf = *(const v2f*)(M  + (nt * 16 + m) * HP_ + k + kb);  // M^T[k][n]
        acc = __builtin_amdgcn_wmma_f32_16x16x4_f32(false, a, false, bf,
                                                    (short)0, acc, false, false);
    }
    float* O = out + (size_t)b * NPIX;
    int rowoff = (lane >> 4) << 3;
#pragma unroll
    for (int i = 0; i < 8; ++i)
        O[(mt * 16 + i + rowoff) * H_ + nt * 16 + m] = acc[i];
}

// ---------------------------------------------------------------------------
// Kernel 4: top-200 mean per batch via two-level LDS histogram over [0,4].
// Values provably in [0,4]: each distance <= 2, blur/upsample convex combos.
// ---------------------------------------------------------------------------
__global__ void topk_kernel(const float* __restrict__ amap,
                            float* __restrict__ score) {
    __shared__ unsigned cnt[4096];
    __shared__ unsigned cnt2[4096];
    __shared__ float    sum2[4096];
    __shared__ float    s_sumhi;
    __shared__ int      s_tbin;
    __shared__ unsigned s_cab;

    int b = blockIdx.x;
    const float* x = amap + (size_t)b * NPIX;

    for (int i = threadIdx.x; i < 4096; i += blockDim.x) {
        cnt[i] = 0u; cnt2[i] = 0u; sum2[i] = 0.f;
    }
    if (threadIdx.x == 0) s_sumhi = 0.f;
    __syncthreads();

    const float scale = 4096.0f / 4.0f;
    for (int i = threadIdx.x; i < NPIX; i += blockDim.x) {
        float v = x[i];
        int bin = (int)(v * scale);
        bin = min(max(bin, 0), 4095);
        atomicAdd(&cnt[bin], 1u);
    }
    __syncthreads();

    if (threadIdx.x == 0) {
        unsigned acc = 0; int tb = 0; unsigned cab = 0;
        for (int i = 4095; i >= 0; --i) {
            unsigned prev = acc;
            acc += cnt[i];
            if (acc >= (unsigned)TOPK) { tb = i; cab = prev; break; }
        }
        s_tbin = tb; s_cab = cab;
    }
    __syncthreads();

    int tb = s_tbin;
    unsigned cab = s_cab;
    float lo = (float)tb / scale;
    float scale2 = 4096.0f * scale;   // sub-bin resolution within one bin
    float local_hi = 0.f;
    for (int i = threadIdx.x; i < NPIX; i += blockDim.x) {
        float v = x[i];
        int bin = (int)(v * scale);
        bin = min(max(bin, 0), 4095);
        if (bin > tb) {
            local_hi += v;
        } else if (bin == tb) {
            int b2 = (int)((v - lo) * scale2);
            b2 = min(max(b2, 0), 4095);
            atomicAdd(&cnt2[b2], 1u);
            atomicAdd(&sum2[b2], v);
        }
    }
    atomicAdd(&s_sumhi, local_hi);
    __syncthreads();

    if (threadIdx.x == 0) {
        unsigned need = (unsigned)TOPK - cab;    // >= 1
        float total = s_sumhi;
        for (int i = 4095; i >= 0 && need > 0; --i) {
            unsigned c = cnt2[i];
            if (c == 0u) continue;
            if (c <= need) { total += sum2[i]; need -= c; }
            else           { total += sum2[i] * ((float)need / (float)c); need = 0; }
        }
        score[b] = total / (float)TOPK;
    }
}

// ---------------------------------------------------------------------------
extern "C" void kernel_launch(void* const* d_in, const int* in_sizes, int n_in,
                              void* d_out, int out_size, void* d_ws, size_t ws_size,
                              hipStream_t stream) {
    (void)in_sizes; (void)n_in; (void)out_size; (void)ws_size;

    const float* mid  = (const float*)d_in[0];
    const float* lst  = (const float*)d_in[1];
    const float* pmid = (const float*)d_in[2];
    const float* plst = (const float*)d_in[3];

    float* out   = (float*)d_out;
    float* amap  = out;                                // 16 * 448 * 448
    float* score = out + (size_t)B_ * NPIX;            // 16

    float* ws   = (float*)d_ws;
    float* comb = ws;                                  // 16*1024
    float* bufA = comb + B_ * N_;                      // 448*32
    float* bufB = bufA + H_ * HP_;                     // 448*32
    float* T    = bufB + H_ * HP_;                     // 16*448*32

    // Phase 1: combined anomaly (memory-bound, ~201 MB)
    anomaly_combined_kernel<<<2048, 256, 0, stream>>>(mid, lst, pmid, plst, comb);

    // Phase 2: build M = (V7^3 V5^5) * U  (448x32)
    compute_u_kernel<<<H_, HP_, 0, stream>>>(bufA);
    float* src = bufA; float* dst = bufB;
    for (int i = 0; i < 5; ++i) {
        blur1d_kernel<<<H_, HP_, 0, stream>>>(src, dst, 5);
        float* tmp = src; src = dst; dst = tmp;
    }
    for (int i = 0; i < 3; ++i) {
        blur1d_kernel<<<H_, HP_, 0, stream>>>(src, dst, 7);
        float* tmp = src; src = dst; dst = tmp;
    }
    // after 8 passes result is back in bufA (== src)

    // Phase 3: anomaly map = M * A * M^T via fp32 WMMA
    gemm1_kernel<<<112,  256, 0, stream>>>(src, comb, T);
    gemm2_kernel<<<1568, 256, 0, stream>>>(T, src, amap);

    // Phase 4: top-200 mean per batch
    topk_kernel<<<B_, 1024, 0, stream>>>(amap, score);
}